// LengthRegulator_87428354278017
// MI455X (gfx1250) — compile-verified
//
#include <hip/hip_runtime.h>
#include <hip/hip_bf16.h>
#include <stdint.h>

// Problem shape (fixed by the reference)
#define B_N   32
#define T_N   512
#define D_N   512
#define MAXF  4096   // MAX_FRAMES
#define NWAVE (T_N / 32)

typedef float v4f __attribute__((ext_vector_type(4)));

// ---------------------------------------------------------------------------
// Kernel 1: per-batch inclusive scan of durations -> frame->token index map.
// One block per batch, 512 threads (16 wave32s).
// Two-level scan: wave32 __shfl_up scan (no barriers) + one-wave scan of the
// 16 wave sums. Then each token scatters its <=7 frames; tail frames get -1.
// ---------------------------------------------------------------------------
__global__ void lr_scan_kernel(const int* __restrict__ dur,
                               int* __restrict__ idx_map,
                               float* __restrict__ mel_lens) {
    const int b    = blockIdx.x;
    const int t    = threadIdx.x;
    const int lane = t & 31;
    const int wid  = t >> 5;

    __shared__ int wsum[NWAVE];

    const int d = dur[(size_t)b * T_N + t];

    // wave32 inclusive scan of d -> v
    int v = d;
    #pragma unroll
    for (int off = 1; off < 32; off <<= 1) {
        int u = __shfl_up(v, off, 32);
        if (lane >= off) v += u;
    }
    if (lane == 31) wsum[wid] = v;   // wave total
    __syncthreads();

    // one wave scans the 16 wave sums (inclusive)
    if (wid == 0 && lane < NWAVE) {
        int s = wsum[lane];
        #pragma unroll
        for (int off = 1; off < NWAVE; off <<= 1) {
            int u = __shfl_up(s, off, 32);
            if (lane >= off) s += u;
        }
        wsum[lane] = s;
    }
    __syncthreads();

    const int base  = (wid > 0) ? wsum[wid - 1] : 0;
    const int end   = v + base;          // ends[t]
    const int start = end - d;           // ends[t-1]
    const int total = wsum[NWAVE - 1];   // total frames (<= 3584 < 4096)

    int* row = idx_map + (size_t)b * MAXF;
    // frames [start, end) map to token t  (searchsorted(..., 'right'))
    for (int f = start; f < end; ++f) row[f] = t;
    // frames >= total are masked to zero in the reference -> sentinel -1
    for (int f = total + t; f < MAXF; f += T_N) row[f] = -1;

    if (t == 0) mel_lens[b] = (float)(total > 0 ? total : 1);
}

// ---------------------------------------------------------------------------
// Kernel 2: one output row (2 KB) per block, 128 threads, 16 B per lane.
// Masked rows: nontemporal b128 zero stores (no reads at all).
// Live rows: gfx1250 async global->LDS->global copy tracked by ASYNCcnt,
// with a non-temporal hint on the store so the 256 MiB streaming output
// does not evict the L2-resident 32 MiB of x (which is re-read ~3.5x).
// Each wave owns its own 512 B LDS slice, so a per-wave s_wait_asynccnt is
// the only synchronization needed between the load and the store.
// ---------------------------------------------------------------------------
__global__ void lr_gather_kernel(const float* __restrict__ x,
                                 const int* __restrict__ idx_map,
                                 float* __restrict__ out) {
    const int p = blockIdx.x;   // frame position 0..4095
    const int b = blockIdx.y;   // batch
    const int c = threadIdx.x;  // 0..127, 4 floats (16 B) per thread

    __shared__ char stage[128 * 16];   // 2 KB row staging

    const int idx = idx_map[(size_t)b * MAXF + p];  // uniform across block

    float* dst = out + (((size_t)b * MAXF + p) * D_N) + (size_t)c * 4;

    if (idx < 0) {
        // masked frame: stream zeros, bypass caches (output >> L2 capacity)
        v4f z = {0.f, 0.f, 0.f, 0.f};
        __builtin_nontemporal_store(z, (v4f*)dst);
        return;
    }

    const float* src = x + (((size_t)b * T_N + idx) * D_N) + (size_t)c * 4;
    // LDS byte offset of this lane's 16 B slot (flat shared addr low 32 bits)
    uint32_t lds_off = (uint32_t)(uintptr_t)(stage + (size_t)c * 16);

    // global -> LDS (ASYNCcnt++); x stays regular-temporal (L2-resident)
    asm volatile("global_load_async_to_lds_b128 %0, %1, off"
                 :: "v"(lds_off), "v"(src) : "memory");
    // wait for this wave's async load to land in LDS
    asm volatile("s_wait_asynccnt 0" ::: "memory");
    // LDS -> global (ASYNCcnt++), non-temporal streaming store
    asm volatile("global_store_async_from_lds_b128 %0, %1, off th:TH_STORE_NT"
                 :: "v"(dst), "v"(lds_off) : "memory");
    asm volatile("s_wait_asynccnt 0" ::: "memory");
}

// ---------------------------------------------------------------------------
// Launch: d_in[0] = x (f32, B*T*D), d_in[1] = durations (i32, B*T).
// d_out = [ out (B*MAXF*D f32) | mel_lens (B, stored as f32 values) ].
// d_ws  = frame->token map, B*MAXF int32 = 512 KB.
// ---------------------------------------------------------------------------
extern "C" void kernel_launch(void* const* d_in, const int* in_sizes, int n_in,
                              void* d_out, int out_size, void* d_ws, size_t ws_size,
                              hipStream_t stream) {
    (void)in_sizes; (void)n_in; (void)out_size; (void)ws_size;

    const float* x   = (const float*)d_in[0];
    const int*   dur = (const int*)d_in[1];

    float* out = (float*)d_out;
    float* mel = out + (size_t)B_N * MAXF * D_N;
    int*   idx_map = (int*)d_ws;

    lr_scan_kernel<<<dim3(B_N), dim3(T_N), 0, stream>>>(dur, idx_map, mel);
    lr_gather_kernel<<<dim3(MAXF, B_N), dim3(D_N / 4), 0, stream>>>(x, idx_map, out);
}